// GCN_8675833938683
// MI455X (gfx1250) — compile-verified
//
#include <hip/hip_runtime.h>
#include <math.h>

#define N_NODES 100000
#define F_IN    128
#define F_MID   16
#define F_OUT   4

typedef __attribute__((ext_vector_type(2))) float v2f;
typedef __attribute__((ext_vector_type(8))) float v8f;

// ---------------- degree / norm ----------------

__global__ void k_deg_init(float* deg) {
    int n = blockIdx.x * blockDim.x + threadIdx.x;
    if (n < N_NODES) deg[n] = 1.0f;  // self-loop
}

__global__ void k_deg_scatter(const int* __restrict__ dst, float* deg, int E) {
    int e = blockIdx.x * blockDim.x + threadIdx.x;
    if (e < E) atomicAdd(&deg[dst[e]], 1.0f);
}

__global__ void k_dinv(float* deg) {
    int n = blockIdx.x * blockDim.x + threadIdx.x;
    if (n < N_NODES) {
        float d = deg[n];
        deg[n] = d > 0.0f ? rsqrtf(d) : 0.0f;
    }
}

// ---------------- layer 1 GEMM: h1 = x @ W1 via V_WMMA_F32_16X16X4_F32 ----------------
// One wave handles a 16-row tile of x (100000 / 16 = 6250 tiles exactly).
// A (16x4 f32): lanes 0-15 hold M=lane K={0,1}; lanes 16-31 hold M=lane-16 K={2,3}.
// B (4x16 f32): lanes 0-15 hold N=lane K={0,1}; lanes 16-31 hold N=lane-16 K={2,3}.
// C/D (16x16 f32, 8 VGPRs): VGPR i -> (M = i + (lane>=16 ? 8:0), N = lane&15).

__global__ void k_gemm1_wmma(const float* __restrict__ x, const float* __restrict__ W1,
                             float* __restrict__ h1, int nTiles) {
    int wave = (blockIdx.x * blockDim.x + threadIdx.x) >> 5;
    int lane = threadIdx.x & 31;
    if (wave >= nTiles) return;           // wave-uniform: EXEC stays all-ones

    int m     = lane & 15;
    int khalf = (lane >> 4) << 1;         // 0 or 2
    int row   = wave * 16 + m;

    v8f c = {};
    #pragma unroll 4
    for (int k0 = 0; k0 < F_IN; k0 += 4) {
        v2f a, b;
        a.x = x[row * F_IN + k0 + khalf];
        a.y = x[row * F_IN + k0 + khalf + 1];
        b.x = W1[(k0 + khalf)     * F_MID + m];
        b.y = W1[(k0 + khalf + 1) * F_MID + m];
        c = __builtin_amdgcn_wmma_f32_16x16x4_f32(false, a, false, b,
                                                  (short)0, c, false, false);
    }

    int mbase = wave * 16 + ((lane >> 4) << 3);
    #pragma unroll
    for (int i = 0; i < 8; ++i)
        h1[(mbase + i) * F_MID + m] = c[i];
}

// ---------------- layer 1 aggregate ----------------

__global__ void k_out1_init(const float* __restrict__ h1, const float* __restrict__ dinv,
                            const float* __restrict__ b1, float* __restrict__ out1) {
    int t = blockIdx.x * blockDim.x + threadIdx.x;
    if (t < N_NODES * F_MID) {
        int n = t >> 4, f = t & 15;
        float di = dinv[n];
        out1[t] = h1[t] * di * di + b1[f];  // self-loop norm = dinv^2
    }
}

__global__ void k_scatter1(const int* __restrict__ src, const int* __restrict__ dst,
                           const float* __restrict__ h1, const float* __restrict__ dinv,
                           float* out1, int E) {
    int t = blockIdx.x * blockDim.x + threadIdx.x;
    if (t < E * F_MID) {
        int e = t >> 4, f = t & 15;
        int s = src[e], d = dst[e];
        float w = dinv[s] * dinv[d];
        atomicAdd(&out1[d * F_MID + f], h1[s * F_MID + f] * w);
    }
}

// ---------------- layer 2 GEMM: h2 = relu(out1) @ W2 via WMMA (N padded 4->16) --------

__global__ void k_gemm2_wmma(const float* __restrict__ out1, const float* __restrict__ W2,
                             float* __restrict__ h2, int nTiles) {
    int wave = (blockIdx.x * blockDim.x + threadIdx.x) >> 5;
    int lane = threadIdx.x & 31;
    if (wave >= nTiles) return;

    int m     = lane & 15;
    int khalf = (lane >> 4) << 1;
    int row   = wave * 16 + m;

    v8f c = {};
    #pragma unroll
    for (int k0 = 0; k0 < F_MID; k0 += 4) {
        v2f a, b;
        float a0 = out1[row * F_MID + k0 + khalf];
        float a1 = out1[row * F_MID + k0 + khalf + 1];
        a.x = a0 > 0.0f ? a0 : 0.0f;      // fused ReLU
        a.y = a1 > 0.0f ? a1 : 0.0f;
        b.x = (m < F_OUT) ? W2[(k0 + khalf)     * F_OUT + m] : 0.0f;
        b.y = (m < F_OUT) ? W2[(k0 + khalf + 1) * F_OUT + m] : 0.0f;
        c = __builtin_amdgcn_wmma_f32_16x16x4_f32(false, a, false, b,
                                                  (short)0, c, false, false);
    }

    int mbase = wave * 16 + ((lane >> 4) << 3);
    if (m < F_OUT) {
        #pragma unroll
        for (int i = 0; i < 8; ++i)
            h2[(mbase + i) * F_OUT + m] = c[i];
    }
}

// ---------------- layer 2 aggregate (into d_out) ----------------

__global__ void k_out2_init(const float* __restrict__ h2, const float* __restrict__ dinv,
                            const float* __restrict__ b2, float* __restrict__ out) {
    int t = blockIdx.x * blockDim.x + threadIdx.x;
    if (t < N_NODES * F_OUT) {
        int n = t >> 2, j = t & 3;
        float di = dinv[n];
        out[t] = h2[t] * di * di + b2[j];
    }
}

__global__ void k_scatter2(const int* __restrict__ src, const int* __restrict__ dst,
                           const float* __restrict__ h2, const float* __restrict__ dinv,
                           float* out, int E) {
    int t = blockIdx.x * blockDim.x + threadIdx.x;
    if (t < E * F_OUT) {
        int e = t >> 2, j = t & 3;
        int s = src[e], d = dst[e];
        float w = dinv[s] * dinv[d];
        atomicAdd(&out[d * F_OUT + j], h2[s * F_OUT + j] * w);
    }
}

// ---------------- log_softmax over 4 classes, in-place ----------------

__global__ void k_log_softmax(float* out) {
    int n = blockIdx.x * blockDim.x + threadIdx.x;
    if (n < N_NODES) {
        float v0 = out[n * 4 + 0], v1 = out[n * 4 + 1];
        float v2 = out[n * 4 + 2], v3 = out[n * 4 + 3];
        float m = fmaxf(fmaxf(v0, v1), fmaxf(v2, v3));
        float s = __expf(v0 - m) + __expf(v1 - m) + __expf(v2 - m) + __expf(v3 - m);
        float l = m + __logf(s);
        out[n * 4 + 0] = v0 - l;
        out[n * 4 + 1] = v1 - l;
        out[n * 4 + 2] = v2 - l;
        out[n * 4 + 3] = v3 - l;
    }
}

// ---------------- launcher ----------------

extern "C" void kernel_launch(void* const* d_in, const int* in_sizes, int n_in,
                              void* d_out, int out_size, void* d_ws, size_t ws_size,
                              hipStream_t stream) {
    const float* x  = (const float*)d_in[0];
    const int*   ei = (const int*)d_in[1];     // [2, E] flattened
    const float* W1 = (const float*)d_in[2];
    const float* b1 = (const float*)d_in[3];
    const float* W2 = (const float*)d_in[4];
    const float* b2 = (const float*)d_in[5];
    float* out = (float*)d_out;

    const int E = in_sizes[1] / 2;
    const int* src = ei;
    const int* dst = ei + E;

    // workspace layout (floats)
    float* ws   = (float*)d_ws;
    float* dinv = ws;                              // N       (deg then dinv in place)
    float* h1   = ws + N_NODES;                    // N * 16
    float* out1 = ws + N_NODES * 17;               // N * 16
    float* h2   = ws + N_NODES * 33;               // N * 4

    const int B = 256;
    auto blocks = [](long long n, int b) { return (int)((n + b - 1) / b); };

    // degree / normalization
    k_deg_init   <<<blocks(N_NODES, B), B, 0, stream>>>(dinv);
    k_deg_scatter<<<blocks(E, B),       B, 0, stream>>>(dst, dinv, E);
    k_dinv       <<<blocks(N_NODES, B), B, 0, stream>>>(dinv);

    // layer 1: WMMA GEMM + self-loop init + edge scatter
    const int nTiles = N_NODES / 16;   // 6250
    k_gemm1_wmma <<<blocks((long long)nTiles * 32, B), B, 0, stream>>>(x, W1, h1, nTiles);
    k_out1_init  <<<blocks((long long)N_NODES * 16, B), B, 0, stream>>>(h1, dinv, b1, out1);
    k_scatter1   <<<blocks((long long)E * 16, B),       B, 0, stream>>>(src, dst, h1, dinv, out1, E);

    // layer 2: fused-ReLU WMMA GEMM + init + edge scatter into d_out
    k_gemm2_wmma <<<blocks((long long)nTiles * 32, B), B, 0, stream>>>(out1, W2, h2, nTiles);
    k_out2_init  <<<blocks((long long)N_NODES * 4, B),  B, 0, stream>>>(h2, dinv, b2, out);
    k_scatter2   <<<blocks((long long)E * 4, B),        B, 0, stream>>>(src, dst, h2, dinv, out, E);

    // log_softmax
    k_log_softmax<<<blocks(N_NODES, B), B, 0, stream>>>(out);
}